// DropoutModel_90426241450316
// MI455X (gfx1250) — compile-verified
//
#include <hip/hip_runtime.h>

#define NQ      15
#define NL      4
#define BATCH   32
#define DIM     (1 << NQ)     // 32768
#define THREADS 1024

typedef float v8f __attribute__((ext_vector_type(8)));
typedef float v2f __attribute__((ext_vector_type(2)));

struct C2 { float re, im; };
__device__ inline C2 cmul(C2 a, C2 b) { return { a.re*b.re - a.im*b.im, a.re*b.im + a.im*b.re }; }
__device__ inline C2 cadd(C2 a, C2 b) { return { a.re + b.re, a.im + b.im }; }

struct M2 { C2 m[4]; };       // row-major 2x2 complex
__device__ inline M2 mmul(const M2& p, const M2& q) {
    M2 r;
    r.m[0] = cadd(cmul(p.m[0], q.m[0]), cmul(p.m[1], q.m[2]));
    r.m[1] = cadd(cmul(p.m[0], q.m[1]), cmul(p.m[1], q.m[3]));
    r.m[2] = cadd(cmul(p.m[2], q.m[0]), cmul(p.m[3], q.m[2]));
    r.m[3] = cadd(cmul(p.m[2], q.m[1]), cmul(p.m[3], q.m[3]));
    return r;
}
__device__ inline M2 mk_ry(float th) {
    float c = __cosf(0.5f * th), s = __sinf(0.5f * th);
    M2 r; r.m[0] = {c,0}; r.m[1] = {-s,0}; r.m[2] = {s,0}; r.m[3] = {c,0}; return r;
}
__device__ inline M2 mk_rz(float th) {
    float c = __cosf(0.5f * th), s = __sinf(0.5f * th);
    M2 r; r.m[0] = {c,-s}; r.m[1] = {0,0}; r.m[2] = {0,0}; r.m[3] = {c,s}; return r;
}
__device__ inline M2 mk_rx(float th) {
    float c = __cosf(0.5f * th), s = __sinf(0.5f * th);
    M2 r; r.m[0] = {c,0}; r.m[1] = {0,-s}; r.m[2] = {0,-s}; r.m[3] = {c,0}; return r;
}

__launch_bounds__(THREADS, 1)
__global__ void qsim_kernel(const float* __restrict__ x,
                            const float* __restrict__ thetas,
                            float* __restrict__ out)
{
    // State (planar complex), fits in 320KB WGP LDS
    __shared__ float sRe[DIM];             // 128 KB
    __shared__ float sIm[DIM];             // 128 KB
    __shared__ float gAre[256], gAim[256]; // G_top  A[m][k] row-major (m*16+k)
    __shared__ float gBre[256], gBim[256]; // G_botT B[k][n] row-major (k*16+n)
    __shared__ float uG[NL * NQ][8];       // fused 2x2 gates
    __shared__ float red[THREADS];

    const int b   = blockIdx.x;
    const int tid = threadIdx.x;

    // ---- init |00...0> ----
    for (int i = tid; i < DIM; i += THREADS) { sRe[i] = 0.f; sIm[i] = 0.f; }
    if (tid == 0) sRe[0] = 1.f;

    // ---- fused per-qubit gate U = RX(t3)*RZ(t2)*RX(t1)*RZ(x2)*RY(x1) ----
    if (tid < NL * NQ) {
        const int l = tid / NQ, q = tid % NQ;
        const float xv = x[b];
        const float x1 = asinf(xv);
        const float x2 = acosf(xv * xv);
        const float t1 = thetas[(l * NQ + q) * 3 + 0];
        const float t2 = thetas[(l * NQ + q) * 3 + 1];
        const float t3 = thetas[(l * NQ + q) * 3 + 2];
        M2 U = mmul(mk_rx(t3), mmul(mk_rz(t2), mmul(mk_rx(t1), mmul(mk_rz(x2), mk_ry(x1)))));
        #pragma unroll
        for (int t = 0; t < 4; ++t) { uG[tid][2*t] = U.m[t].re; uG[tid][2*t+1] = U.m[t].im; }
    }
    __syncthreads();

    const int lane = tid & 31;
    const int wave = tid >> 5;
    const int half = lane >> 4;      // 0: lanes 0-15, 1: lanes 16-31
    const int nlo  = lane & 15;

    for (int l = 0; l < NL; ++l) {
        // ---- build G_top (qubits 0-3) and G_botT (qubits 11-14) ----
        if (tid < 512) {
            const int e = tid & 255;
            C2 p = {1.f, 0.f};
            if (tid < 256) {
                const int m = e >> 4, k = e & 15;
                #pragma unroll
                for (int j = 0; j < 4; ++j) {
                    const int bo = (m >> (3 - j)) & 1, bi = (k >> (3 - j)) & 1;
                    const float* u = uG[l * NQ + j];
                    C2 f = { u[(bo * 2 + bi) * 2], u[(bo * 2 + bi) * 2 + 1] };
                    p = cmul(p, f);
                }
                gAre[e] = p.re; gAim[e] = p.im;            // A[m][k]
            } else {
                const int k = e >> 4, n = e & 15;          // B[k][n] = G_bot[n][k]
                #pragma unroll
                for (int j = 0; j < 4; ++j) {
                    const int bo = (n >> (3 - j)) & 1, bi = (k >> (3 - j)) & 1;
                    const float* u = uG[l * NQ + 11 + j];
                    C2 f = { u[(bo * 2 + bi) * 2], u[(bo * 2 + bi) * 2 + 1] };
                    p = cmul(p, f);
                }
                gBre[e] = p.re; gBim[e] = p.im;
            }
        }
        __syncthreads();

        // ---- top group (qubits 0-3): S(16x2048) = G_top * S, WMMA f32 16x16x4 ----
        {
            v2f Ar[4], Ai[4];
            #pragma unroll
            for (int c = 0; c < 4; ++c) {
                const int k0 = 4 * c + 2 * half;
                Ar[c].x = gAre[nlo * 16 + k0]; Ar[c].y = gAre[nlo * 16 + k0 + 1];
                Ai[c].x = gAim[nlo * 16 + k0]; Ai[c].y = gAim[nlo * 16 + k0 + 1];
            }
            #pragma unroll 1
            for (int it = 0; it < 4; ++it) {
                const int colBase = (wave + 32 * it) * 16;
                v8f accRe = {}; v8f accIm = {};
                #pragma unroll
                for (int c = 0; c < 4; ++c) {
                    const int k0 = 4 * c + 2 * half;
                    v2f Br, Bi, BiN;
                    Br.x = sRe[k0 * 2048 + colBase + nlo]; Br.y = sRe[(k0 + 1) * 2048 + colBase + nlo];
                    Bi.x = sIm[k0 * 2048 + colBase + nlo]; Bi.y = sIm[(k0 + 1) * 2048 + colBase + nlo];
                    BiN.x = -Bi.x; BiN.y = -Bi.y;
                    accRe = __builtin_amdgcn_wmma_f32_16x16x4_f32(false, Ar[c], false, Br,  (short)0, accRe, false, false);
                    accRe = __builtin_amdgcn_wmma_f32_16x16x4_f32(false, Ai[c], false, BiN, (short)0, accRe, false, false);
                    accIm = __builtin_amdgcn_wmma_f32_16x16x4_f32(false, Ar[c], false, Bi,  (short)0, accIm, false, false);
                    accIm = __builtin_amdgcn_wmma_f32_16x16x4_f32(false, Ai[c], false, Br,  (short)0, accIm, false, false);
                }
                #pragma unroll
                for (int r = 0; r < 8; ++r) {
                    const int m = r + 8 * half;
                    sRe[m * 2048 + colBase + nlo] = accRe[r];
                    sIm[m * 2048 + colBase + nlo] = accIm[r];
                }
            }
        }
        __syncthreads();

        // ---- bottom group (qubits 11-14): S(2048x16) = S * G_botT, WMMA ----
        {
            v2f Br[4], Bi[4];
            #pragma unroll
            for (int c = 0; c < 4; ++c) {
                const int k0 = 4 * c + 2 * half;
                Br[c].x = gBre[k0 * 16 + nlo]; Br[c].y = gBre[(k0 + 1) * 16 + nlo];
                Bi[c].x = gBim[k0 * 16 + nlo]; Bi[c].y = gBim[(k0 + 1) * 16 + nlo];
            }
            #pragma unroll 1
            for (int it = 0; it < 4; ++it) {
                const int rowBase = (wave + 32 * it) * 16;
                const int arow = rowBase + nlo;
                v8f accRe = {}; v8f accIm = {};
                #pragma unroll
                for (int c = 0; c < 4; ++c) {
                    const int k0 = 4 * c + 2 * half;
                    v2f Ar, Ai, AiN;
                    Ar.x = sRe[arow * 16 + k0]; Ar.y = sRe[arow * 16 + k0 + 1];
                    Ai.x = sIm[arow * 16 + k0]; Ai.y = sIm[arow * 16 + k0 + 1];
                    AiN.x = -Ai.x; AiN.y = -Ai.y;
                    accRe = __builtin_amdgcn_wmma_f32_16x16x4_f32(false, Ar,  false, Br[c], (short)0, accRe, false, false);
                    accRe = __builtin_amdgcn_wmma_f32_16x16x4_f32(false, AiN, false, Bi[c], (short)0, accRe, false, false);
                    accIm = __builtin_amdgcn_wmma_f32_16x16x4_f32(false, Ar,  false, Bi[c], (short)0, accIm, false, false);
                    accIm = __builtin_amdgcn_wmma_f32_16x16x4_f32(false, Ai,  false, Br[c], (short)0, accIm, false, false);
                }
                #pragma unroll
                for (int r = 0; r < 8; ++r) {
                    const int row = rowBase + r + 8 * half;
                    sRe[row * 16 + nlo] = accRe[r];
                    sIm[row * 16 + nlo] = accIm[r];
                }
            }
        }
        __syncthreads();

        // ---- middle qubits 4..10: fused scalar 2x2 passes ----
        for (int q = 4; q <= 10; ++q) {
            const int s = 14 - q;
            const int stride = 1 << s;
            const float* u = uG[l * NQ + q];
            const float ar = u[0], ai = u[1], br = u[2], bi = u[3];
            const float cr = u[4], ci = u[5], dr = u[6], di = u[7];
            #pragma unroll 4
            for (int p = tid; p < DIM / 2; p += THREADS) {
                const int i0 = ((p >> s) << (s + 1)) | (p & (stride - 1));
                const int i1 = i0 | stride;
                const float r0 = sRe[i0], m0 = sIm[i0], r1 = sRe[i1], m1 = sIm[i1];
                sRe[i0] = ar * r0 - ai * m0 + br * r1 - bi * m1;
                sIm[i0] = ar * m0 + ai * r0 + br * m1 + bi * r1;
                sRe[i1] = cr * r0 - ci * m0 + dr * r1 - di * m1;
                sIm[i1] = cr * m0 + ci * r0 + dr * m1 + di * r1;
            }
            __syncthreads();
        }

        // ---- CNOT chain == Gray-code permutation: amp'[J] = amp[J ^ (J>>1)] ----
        // Split into Re round then Im round to halve register pressure.
        {
            float t[32];
            #pragma unroll
            for (int e = 0; e < 32; ++e) {
                const int J = tid + THREADS * e;
                t[e] = sRe[J ^ (J >> 1)];
            }
            __syncthreads();
            #pragma unroll
            for (int e = 0; e < 32; ++e) sRe[tid + THREADS * e] = t[e];
            __syncthreads();
            #pragma unroll
            for (int e = 0; e < 32; ++e) {
                const int J = tid + THREADS * e;
                t[e] = sIm[J ^ (J >> 1)];
            }
            __syncthreads();
            #pragma unroll
            for (int e = 0; e < 32; ++e) sIm[tid + THREADS * e] = t[e];
        }
        __syncthreads();
    }

    // ---- <Z^15> = sum_k (-1)^popcount(k) |amp_k|^2 ----
    float acc = 0.f;
    for (int i = tid; i < DIM; i += THREADS) {
        const float p = sRe[i] * sRe[i] + sIm[i] * sIm[i];
        acc += (__popc(i) & 1) ? -p : p;
    }
    red[tid] = acc;
    __syncthreads();
    for (int off = THREADS / 2; off > 0; off >>= 1) {
        if (tid < off) red[tid] += red[tid + off];
        __syncthreads();
    }
    if (tid == 0) out[b] = red[0];
}

extern "C" void kernel_launch(void* const* d_in, const int* in_sizes, int n_in,
                              void* d_out, int out_size, void* d_ws, size_t ws_size,
                              hipStream_t stream) {
    (void)in_sizes; (void)n_in; (void)out_size; (void)d_ws; (void)ws_size;
    const float* x      = (const float*)d_in[0];   // (32,)
    const float* thetas = (const float*)d_in[1];   // (4,15,3)
    float* out = (float*)d_out;                    // (32,) float32
    qsim_kernel<<<dim3(BATCH), dim3(THREADS), 0, stream>>>(x, thetas, out);
}